// Model_63840393887771
// MI455X (gfx1250) — compile-verified
//
#include <hip/hip_runtime.h>
#include <math.h>

// Problem constants (from reference)
#define D_MODEL  1024
#define D_STATE  16
#define D_CONV   4
#define D_INNER  2048
#define DT_RANK  64
#define BATCH    2
#define SEQLEN   2048
#define BL       (BATCH * SEQLEN)        // 4096 rows for all GEMMs
#define XDBL_LD  (DT_RANK + 2 * D_STATE) // 96

typedef __attribute__((ext_vector_type(2))) float v2f;
typedef __attribute__((ext_vector_type(4))) float v4f;
typedef __attribute__((ext_vector_type(8))) float v8f;

__device__ __forceinline__ float silu_f(float v)     { return v / (1.f + expf(-v)); }
__device__ __forceinline__ float softplus_f(float v) { return (v > 20.f) ? v : log1pf(expf(v)); }

// ---------------------------------------------------------------------------
// C[m][n] = sum_k A[m][k] * W[n][k]
//   A: M x K row-major (stride lda), W: N x K row-major (stride ldw)
// Block = 256 threads = 8 waves. Block tile: 128 (M) x 16*NT (N).
// K tiled by KC=32, staged through LDS (padded stride 36 -> bank-conflict-free
// fragment reads, 16B-aligned b128 stores), with global->reg prefetch of the
// next chunk overlapping the WMMA work of the current chunk.
// EPI: 0 = none, 1 = softplus(val + bias[n])
// ---------------------------------------------------------------------------
template <int NT, int EPI>
__launch_bounds__(256)
__global__ void gemm_wmma_f32(const float* __restrict__ A, int lda,
                              const float* __restrict__ W, int ldw,
                              float* __restrict__ C, int ldc, int K,
                              const float* __restrict__ bias)
{
    constexpr int KC      = 32;          // K-chunk
    constexpr int KC4     = KC / 4;      // v4f per row
    constexpr int LDS_LD  = KC + 4;      // 36 floats: 16B-aligned rows, no bank conflicts
    constexpr int NROWS_B = 16 * NT;
    constexpr int NA      = (128 * KC4) / 256;      // A v4f per thread (=4)
    constexpr int NB      = (NROWS_B * KC4) / 256;  // B v4f per thread (4 or 3)

    __shared__ float As[128 * LDS_LD];
    __shared__ float Bs[NROWS_B * LDS_LD];

    const int tid  = threadIdx.x;
    const int wave = tid >> 5;
    const int lane = tid & 31;
    const int half = lane >> 4;   // K-pair select (A/B), M-half select (C/D)
    const int lidx = lane & 15;
    const int m_blk  = blockIdx.x * 128;
    const int n_base = blockIdx.y * NROWS_B;

    v8f acc[NT];
#pragma unroll
    for (int t = 0; t < NT; ++t) acc[t] = (v8f){0.f,0.f,0.f,0.f,0.f,0.f,0.f,0.f};

    v4f ra[NA], rb[NB];

    // ---- prologue: fetch chunk 0 into registers (global_load_b128) ----
#pragma unroll
    for (int i = 0; i < NA; ++i) {
        const int idx = tid + i * 256, r = idx >> 3, c = (idx & 7) * 4;
        ra[i] = *(const v4f*)(A + (size_t)(m_blk + r) * lda + c);
    }
#pragma unroll
    for (int i = 0; i < NB; ++i) {
        const int idx = tid + i * 256, r = idx >> 3, c = (idx & 7) * 4;
        rb[i] = *(const v4f*)(W + (size_t)(n_base + r) * ldw + c);
    }

    for (int k0 = 0; k0 < K; k0 += KC) {
        // ---- stage current chunk to LDS ----
#pragma unroll
        for (int i = 0; i < NA; ++i) {
            const int idx = tid + i * 256, r = idx >> 3, c = (idx & 7) * 4;
            *(v4f*)(As + r * LDS_LD + c) = ra[i];
        }
#pragma unroll
        for (int i = 0; i < NB; ++i) {
            const int idx = tid + i * 256, r = idx >> 3, c = (idx & 7) * 4;
            *(v4f*)(Bs + r * LDS_LD + c) = rb[i];
        }
        __syncthreads();

        // ---- prefetch next chunk (overlaps with WMMA below) ----
        if (k0 + KC < K) {
#pragma unroll
            for (int i = 0; i < NA; ++i) {
                const int idx = tid + i * 256, r = idx >> 3, c = (idx & 7) * 4;
                ra[i] = *(const v4f*)(A + (size_t)(m_blk + r) * lda + (k0 + KC) + c);
            }
#pragma unroll
            for (int i = 0; i < NB; ++i) {
                const int idx = tid + i * 256, r = idx >> 3, c = (idx & 7) * 4;
                rb[i] = *(const v4f*)(W + (size_t)(n_base + r) * ldw + (k0 + KC) + c);
            }
        }

        // ---- compute: 8 WMMA K-steps from LDS ----
        const float* a_lds = As + (wave * 16 + lidx) * LDS_LD + 2 * half;
        const float* b_lds = Bs + lidx * LDS_LD + 2 * half;
#pragma unroll
        for (int kk = 0; kk < KC; kk += 4) {
            v2f a = *(const v2f*)(a_lds + kk);
#pragma unroll
            for (int t = 0; t < NT; ++t) {
                v2f b = *(const v2f*)(b_lds + t * 16 * LDS_LD + kk);
                acc[t] = __builtin_amdgcn_wmma_f32_16x16x4_f32(
                    false, a, false, b, (short)0, acc[t], false, false);
            }
        }
        __syncthreads();
    }

    // ---- epilogue: D layout: VGPR v, lanes 0-15 -> M=v, 16-31 -> M=8+v ----
#pragma unroll
    for (int t = 0; t < NT; ++t) {
        const int n = n_base + t * 16 + lidx;
#pragma unroll
        for (int v = 0; v < 8; ++v) {
            const int m = m_blk + wave * 16 + half * 8 + v;
            float val = acc[t][v];
            if (EPI == 1) val = softplus_f(val + bias[n]);
            C[(size_t)m * ldc + n] = val;
        }
    }
}

// ---------------------------------------------------------------------------
// Depthwise causal conv (K=4) + bias + SiLU. One thread per (b,l,d).
// ---------------------------------------------------------------------------
__global__ void conv_silu_kernel(const float* __restrict__ x,
                                 const float* __restrict__ w,
                                 const float* __restrict__ cb,
                                 float* __restrict__ xc)
{
    const int idx = blockIdx.x * blockDim.x + threadIdx.x;
    if (idx >= BATCH * SEQLEN * D_INNER) return;
    const int d = idx % D_INNER;
    const int l = (idx / D_INNER) % SEQLEN;
    const int b = idx / (D_INNER * SEQLEN);
    float acc = cb[d];
#pragma unroll
    for (int k = 0; k < D_CONV; ++k) {
        const int ls = l + k - (D_CONV - 1);
        if (ls >= 0)
            acc += x[((size_t)b * SEQLEN + ls) * D_INNER + d] * w[d * D_CONV + k];
    }
    xc[idx] = silu_f(acc);
}

// ---------------------------------------------------------------------------
// Selective scan. One lane per (b, d, n): 65536 threads = 2048 waves.
// 16-lane butterfly reduction (ds_swizzle path) per step for y = sum_n h*C.
// Writes gated output (y + D*u)*silu(z) in place over the delta buffer.
// ---------------------------------------------------------------------------
__global__ void scan_kernel(const float* delta,             // aliases yout
                            const float* __restrict__ xc,
                            const float* __restrict__ xdbl, // B at col 64, C at 80
                            const float* __restrict__ A_log,
                            const float* __restrict__ Dp,
                            const float* __restrict__ z,
                            float* yout)                    // aliases delta
{
    const int gid = blockIdx.x * blockDim.x + threadIdx.x;
    const int n  = gid & (D_STATE - 1);
    const int ch = gid >> 4;                 // (b, d) channel
    if (ch >= BATCH * D_INNER) return;
    const int d = ch % D_INNER;
    const int b = ch / D_INNER;

    const float An = -expf(A_log[d * D_STATE + n]);
    const float Dd = Dp[d];
    float h = 0.f;

    const size_t row0 = (size_t)b * SEQLEN;
    for (int l = 0; l < SEQLEN; ++l) {
        const size_t off = (row0 + l) * D_INNER + d;
        const float dt = delta[off];
        const float u  = xc[off];
        const float* bc = xdbl + (row0 + l) * XDBL_LD;
        const float Bn = bc[DT_RANK + n];
        const float Cn = bc[DT_RANK + D_STATE + n];

        const float dA = expf(dt * An);
        h = h * dA + (dt * u) * Bn;
        float t = h * Cn;
        t += __shfl_xor(t, 1, 16);
        t += __shfl_xor(t, 2, 16);
        t += __shfl_xor(t, 4, 16);
        t += __shfl_xor(t, 8, 16);
        if (n == 0) {
            const float zz = z[off];
            yout[off] = (t + Dd * u) * silu_f(zz);
        }
    }
}

// ---------------------------------------------------------------------------
extern "C" void kernel_launch(void* const* d_in, const int* in_sizes, int n_in,
                              void* d_out, int out_size, void* d_ws, size_t ws_size,
                              hipStream_t stream)
{
    const float* hs        = (const float*)d_in[0]; // (B,L,D_MODEL)
    const float* in_proj_w = (const float*)d_in[1]; // (2*D_INNER, D_MODEL)
    const float* conv_w    = (const float*)d_in[2]; // (D_INNER,1,4)
    const float* conv_b    = (const float*)d_in[3]; // (D_INNER)
    const float* x_proj_w  = (const float*)d_in[4]; // (96, D_INNER)
    const float* dt_proj_w = (const float*)d_in[5]; // (D_INNER, 64)
    const float* dt_proj_b = (const float*)d_in[6]; // (D_INNER)
    const float* A_log     = (const float*)d_in[7]; // (D_INNER, 16)
    const float* Dp        = (const float*)d_in[8]; // (D_INNER)
    const float* out_w     = (const float*)d_in[9]; // (D_MODEL, D_INNER)
    float* out = (float*)d_out;

    // Workspace layout (bytes)
    const size_t SZ_BLD = (size_t)BL * D_INNER * sizeof(float);      // 32 MB
    char* ws = (char*)d_ws;
    float* x     = (float*)(ws);                      // post in_proj (x half)
    float* z     = (float*)(ws + SZ_BLD);             // post in_proj (z half)
    float* xc    = (float*)(ws + 2 * SZ_BLD);         // post conv+silu
    float* delta = (float*)(ws + 3 * SZ_BLD);         // softplus(...); reused as y
    float* xdbl  = (float*)(ws + 4 * SZ_BLD);         // (BL, 96)

    const dim3 blk(256);

    // 1) in_proj: x = hs @ Wx^T , z = hs @ Wz^T   (M=4096, K=1024, N=2048 each)
    gemm_wmma_f32<8, 0><<<dim3(BL / 128, D_INNER / 128), blk, 0, stream>>>(
        hs, D_MODEL, in_proj_w, D_MODEL, x, D_INNER, D_MODEL, nullptr);
    gemm_wmma_f32<8, 0><<<dim3(BL / 128, D_INNER / 128), blk, 0, stream>>>(
        hs, D_MODEL, in_proj_w + (size_t)D_INNER * D_MODEL, D_MODEL,
        z, D_INNER, D_MODEL, nullptr);

    // 2) depthwise conv + SiLU
    const int nelem = BATCH * SEQLEN * D_INNER;
    conv_silu_kernel<<<(nelem + 255) / 256, blk, 0, stream>>>(x, conv_w, conv_b, xc);

    // 3) x_dbl = xc @ x_proj_w^T  (M=4096, K=2048, N=96 -> NT=6, one N strip)
    gemm_wmma_f32<6, 0><<<dim3(BL / 128, 1), blk, 0, stream>>>(
        xc, D_INNER, x_proj_w, D_INNER, xdbl, XDBL_LD, D_INNER, nullptr);

    // 4) delta = softplus(dt @ dt_proj_w^T + b)  (A = xdbl cols 0..63, lda=96)
    gemm_wmma_f32<8, 1><<<dim3(BL / 128, D_INNER / 128), blk, 0, stream>>>(
        xdbl, XDBL_LD, dt_proj_w, DT_RANK, delta, D_INNER, DT_RANK, dt_proj_b);

    // 5) selective scan + D*u skip + silu(z) gating (writes y over delta)
    const int scan_threads = BATCH * D_INNER * D_STATE; // 65536
    scan_kernel<<<(scan_threads + 255) / 256, blk, 0, stream>>>(
        delta, xc, xdbl, A_log, Dp, z, delta);

    // 6) out = y @ out_proj_w^T  (M=4096, K=2048, N=1024)
    gemm_wmma_f32<8, 0><<<dim3(BL / 128, D_MODEL / 128), blk, 0, stream>>>(
        delta, D_INNER, out_w, D_INNER, out, D_MODEL, D_INNER, nullptr);
}